// LNN_65000035058519
// MI455X (gfx1250) — compile-verified
//
#include <hip/hip_runtime.h>
#include <hip/hip_bf16.h>
#include <math.h>

// ---------------------------------------------------------------------------
// LNN dynamics: grad + 16 Hessian-vector products + 16x16 solve, all GEMM-ified.
// All matrix math runs on V_WMMA_F32_16X16X4_F32 (fp32 exact, required for
// second derivatives). Weights (33.5MB) live in the 192MB L2; activations are
// chunked (64 samples -> 1024 tangent columns) to keep workspace ~53MB.
// GEMM K-loop is double-buffered (LDS ping-pong, one barrier per K-step,
// global fetch for step k+1 overlapped with the 16 WMMAs of step k).
// ---------------------------------------------------------------------------

#define H      2048
#define BATCH  256
#define QD     16
#define ZD     32
#define CHUNK_B 64
#define NCOLS  (CHUNK_B * QD)   // 1024 tangent columns per chunk

typedef float v2f __attribute__((ext_vector_type(2)));
typedef float v8f __attribute__((ext_vector_type(8)));

__device__ __forceinline__ v8f wmma_f32(v2f a, v2f b, v8f c) {
  // D = A(16x4) * B(4x16) + C(16x16), fp32
  return __builtin_amdgcn_wmma_f32_16x16x4_f32(false, a, false, b, (short)0, c,
                                               false, false);
}

// ---------------------------------------------------------------------------
// Tiled GEMM:  C = op(A)(MxK) * X(KxN), stored transposed: Ct (N x M) row-major.
// X is provided transposed: Xt (N x K) row-major (activations are sample-major).
// TA=true: A pointer is (K x M) row-major (i.e., op(A)=A^T of a row-major W).
// EPI: 0 = plain store O0
//      1 = +bias; O0 = softplus(v), O1 = sigmoid(v)
//      2 = +bias; O0 = sigmoid(v),  O1 = sigmoid(v) * aux[m]      (s3 = sg3*w_out)
//      3 = O0 = v, O1 = aux[n*M+m] * v                            (s = sigma (*) u)
// ---------------------------------------------------------------------------
template <bool TA, int EPI>
__global__ __launch_bounds__(128) void gemm_k(
    const float* __restrict__ A, int lda, const float* __restrict__ Xt,
    float* __restrict__ O0, float* __restrict__ O1,
    const float* __restrict__ bias, const float* __restrict__ aux,
    int M, int K, int N) {
  __shared__ float As[2][64 * 20];  // [buf][BM][BK+4] pad -> 16B float4 rows
  __shared__ float Bs[2][64 * 20];  // [buf][BN][BK+4]

  const int tid  = threadIdx.x;
  const int lane = tid & 31;
  const int wave = tid >> 5;        // 4 waves
  const int wm = wave & 1, wn = wave >> 1;
  const int lo = lane & 15, hi = lane >> 4;
  const int n0 = blockIdx.x * 64;
  const int m0 = blockIdx.y * 64;

  v8f acc00 = {0.f,0.f,0.f,0.f,0.f,0.f,0.f,0.f};
  v8f acc01 = {0.f,0.f,0.f,0.f,0.f,0.f,0.f,0.f};
  v8f acc10 = {0.f,0.f,0.f,0.f,0.f,0.f,0.f,0.f};
  v8f acc11 = {0.f,0.f,0.f,0.f,0.f,0.f,0.f,0.f};

  float4 ra0, ra1, rb0, rb1;  // in-flight global fetch (next K-tile)

  // global -> registers for K-tile starting at k0
  auto fetch = [&](int k0) {
    if (!TA) {
      int mm = tid >> 2, kk = (tid & 3) << 2;
      ra0 = make_float4(0.f, 0.f, 0.f, 0.f);
      if (m0 + mm < M)
        ra0 = *(const float4*)(A + (size_t)(m0 + mm) * lda + (k0 + kk));
      int idx = tid + 128;
      mm = idx >> 2; kk = (idx & 3) << 2;
      ra1 = make_float4(0.f, 0.f, 0.f, 0.f);
      if (m0 + mm < M)
        ra1 = *(const float4*)(A + (size_t)(m0 + mm) * lda + (k0 + kk));
    } else {
      int kk = tid >> 4, mm = (tid & 15) << 2;
      ra0 = make_float4(0.f, 0.f, 0.f, 0.f);
      if (m0 + mm < M)
        ra0 = *(const float4*)(A + (size_t)(k0 + kk) * lda + (m0 + mm));
      int idx = tid + 128;
      kk = idx >> 4; mm = (idx & 15) << 2;
      ra1 = make_float4(0.f, 0.f, 0.f, 0.f);
      if (m0 + mm < M)
        ra1 = *(const float4*)(A + (size_t)(k0 + kk) * lda + (m0 + mm));
    }
    {
      int nn = tid >> 2, kk = (tid & 3) << 2;
      rb0 = *(const float4*)(Xt + (size_t)(n0 + nn) * K + (k0 + kk));
      int idx = tid + 128;
      nn = idx >> 2; kk = (idx & 3) << 2;
      rb1 = *(const float4*)(Xt + (size_t)(n0 + nn) * K + (k0 + kk));
    }
  };

  // registers -> LDS buffer buf
  auto stage = [&](int buf) {
    float* as = As[buf];
    float* bs = Bs[buf];
    if (!TA) {
      int mm = tid >> 2, kk = (tid & 3) << 2;
      *(float4*)(&as[mm * 20 + kk]) = ra0;
      int idx = tid + 128;
      mm = idx >> 2; kk = (idx & 3) << 2;
      *(float4*)(&as[mm * 20 + kk]) = ra1;
    } else {
      int kk = tid >> 4, mm = (tid & 15) << 2;
      as[(mm + 0) * 20 + kk] = ra0.x;
      as[(mm + 1) * 20 + kk] = ra0.y;
      as[(mm + 2) * 20 + kk] = ra0.z;
      as[(mm + 3) * 20 + kk] = ra0.w;
      int idx = tid + 128;
      kk = idx >> 4; mm = (idx & 15) << 2;
      as[(mm + 0) * 20 + kk] = ra1.x;
      as[(mm + 1) * 20 + kk] = ra1.y;
      as[(mm + 2) * 20 + kk] = ra1.z;
      as[(mm + 3) * 20 + kk] = ra1.w;
    }
    int nn = tid >> 2, kk = (tid & 3) << 2;
    *(float4*)(&bs[nn * 20 + kk]) = rb0;
    int idx = tid + 128;
    nn = idx >> 2; kk = (idx & 3) << 2;
    *(float4*)(&bs[nn * 20 + kk]) = rb1;
  };

  const int kIters = K >> 4;
  fetch(0);
  stage(0);
  __syncthreads();

  for (int kt = 0; kt < kIters; ++kt) {
    const int cur = kt & 1;
    if (kt + 1 < kIters) fetch((kt + 1) << 4);  // overlap with WMMAs below

    const float* as = As[cur];
    const float* bs = Bs[cur];
    // 4 sub-steps of K=4; A frag: lanes0-15 K=(kk,kk+1), lanes16-31 K=(kk+2,kk+3)
#pragma unroll
    for (int s = 0; s < 4; ++s) {
      int kk = (s << 2) + (hi << 1);
      v2f a0 = *(const v2f*)(&as[(wm * 32 + lo) * 20 + kk]);
      v2f a1 = *(const v2f*)(&as[(wm * 32 + 16 + lo) * 20 + kk]);
      v2f b0 = *(const v2f*)(&bs[(wn * 32 + lo) * 20 + kk]);
      v2f b1 = *(const v2f*)(&bs[(wn * 32 + 16 + lo) * 20 + kk]);
      acc00 = wmma_f32(a0, b0, acc00);
      acc01 = wmma_f32(a0, b1, acc01);
      acc10 = wmma_f32(a1, b0, acc10);
      acc11 = wmma_f32(a1, b1, acc11);
    }

    if (kt + 1 < kIters) stage(cur ^ 1);  // buf cur^1 was last read at kt-1
    __syncthreads();
  }

#pragma unroll
  for (int mi = 0; mi < 2; ++mi) {
#pragma unroll
    for (int ni = 0; ni < 2; ++ni) {
      v8f accv = mi ? (ni ? acc11 : acc10) : (ni ? acc01 : acc00);
      int n  = n0 + wn * 32 + ni * 16 + lo;          // N always multiple of 64
      int mb = m0 + wm * 32 + mi * 16 + hi * 8;      // rows mb..mb+7
      if (mb < M) {
        float vals[8];
#pragma unroll
        for (int v = 0; v < 8; ++v) {
          float val = accv[v];
          int m = mb + v;
          if (EPI == 1) {
            val += bias[m];
            float sg = 1.f / (1.f + expf(-val));
            O1[(size_t)n * M + m] = sg;
            val = fmaxf(val, 0.f) + log1pf(expf(-fabsf(val)));  // softplus
          } else if (EPI == 2) {
            val += bias[m];
            float sg = 1.f / (1.f + expf(-val));
            O1[(size_t)n * M + m] = sg * aux[m];
            val = sg;
          } else if (EPI == 3) {
            O1[(size_t)n * M + m] = aux[(size_t)n * M + m] * val;
          }
          vals[v] = val;
        }
        *(float4*)(O0 + (size_t)n * M + mb) =
            make_float4(vals[0], vals[1], vals[2], vals[3]);
        *(float4*)(O0 + (size_t)n * M + mb + 4) =
            make_float4(vals[4], vals[5], vals[6], vals[7]);
      }
    }
  }
}

// ---------------------------------------------------------------------------
// Elementwise kernels (tangent phase). id = c*H + i, c = chunk-local column,
// b = c0 + c/16 (sample), t = c&15 (direction).
// ---------------------------------------------------------------------------
__global__ __launch_bounds__(256) void k_prep_z(const float* __restrict__ q,
                                                const float* __restrict__ qd,
                                                float* __restrict__ zT) {
  int id = blockIdx.x * 256 + threadIdx.x;
  if (id >= BATCH * ZD) return;
  int n = id >> 5, k = id & 31;
  zT[id] = (k < QD) ? q[n * QD + k] : qd[n * QD + (k - QD)];
}

__global__ __launch_bounds__(256) void k_dx1(const float* __restrict__ sg1,
                                             const float* __restrict__ W_in,
                                             float* __restrict__ dx1, int c0) {
  int id = blockIdx.x * 256 + threadIdx.x;
  int c = id >> 11, i = id & (H - 1);
  int b = c0 + (c >> 4), t = c & 15;
  dx1[id] = sg1[(size_t)b * H + i] * W_in[i * ZD + QD + t];
}

__global__ __launch_bounds__(256) void k_dx2(const float* __restrict__ sg2,
                                             const float* __restrict__ da2,
                                             float* __restrict__ dx2, int c0) {
  int id = blockIdx.x * 256 + threadIdx.x;
  int c = id >> 11, i = id & (H - 1);
  int b = c0 + (c >> 4);
  dx2[id] = sg2[(size_t)b * H + i] * da2[id];
}

__global__ __launch_bounds__(256) void k_ds3(const float* __restrict__ sg3,
                                             const float* __restrict__ wout,
                                             float* da3, int c0) {
  int id = blockIdx.x * 256 + threadIdx.x;
  int c = id >> 11, i = id & (H - 1);
  int b = c0 + (c >> 4);
  float s = sg3[(size_t)b * H + i];
  da3[id] = s * (1.f - s) * wout[i] * da3[id];
}

__global__ __launch_bounds__(256) void k_ds2(const float* __restrict__ sg2,
                                             const float* __restrict__ u2,
                                             const float* __restrict__ du2,
                                             float* da2io, int c0) {
  int id = blockIdx.x * 256 + threadIdx.x;
  int c = id >> 11, i = id & (H - 1);
  size_t bi = (size_t)(c0 + (c >> 4)) * H + i;
  float s = sg2[bi];
  da2io[id] = s * (1.f - s) * u2[bi] * da2io[id] + s * du2[id];
}

__global__ __launch_bounds__(256) void k_ds1(const float* __restrict__ sg1,
                                             const float* __restrict__ u1,
                                             const float* __restrict__ W_in,
                                             float* du1io, int c0) {
  int id = blockIdx.x * 256 + threadIdx.x;
  int c = id >> 11, i = id & (H - 1);
  int t = c & 15;
  size_t bi = (size_t)(c0 + (c >> 4)) * H + i;
  float s = sg1[bi];
  du1io[id] = s * (1.f - s) * u1[bi] * W_in[i * ZD + QD + t] + s * du1io[id];
}

// ---------------------------------------------------------------------------
// Per-sample 16x16 solve: A[t1][t2] = H[16+t1][16+t2] = hvT[(b*16+t2)*32 + 16+t1]
// rhs[t1] = grad[t1] - sum_t2 H[t1][16+t2] * qdot[t2]; LU w/ partial pivoting.
// ---------------------------------------------------------------------------
__global__ __launch_bounds__(64) void k_solve(const float* __restrict__ hvT,
                                              const float* __restrict__ gradT,
                                              const float* __restrict__ qdot,
                                              float* __restrict__ out) {
  int b = blockIdx.x * blockDim.x + threadIdx.x;
  if (b >= BATCH) return;
  float Am[16][16], rhs[16];
#pragma unroll
  for (int t1 = 0; t1 < 16; ++t1) rhs[t1] = gradT[b * ZD + t1];
  for (int t2 = 0; t2 < 16; ++t2) {
    const float* col = hvT + ((size_t)b * 16 + t2) * ZD;
    float qv = qdot[b * 16 + t2];
    for (int t1 = 0; t1 < 16; ++t1) {
      Am[t1][t2] = col[QD + t1];
      rhs[t1] -= col[t1] * qv;
    }
  }
  for (int k = 0; k < 16; ++k) {
    int p = k;
    float best = fabsf(Am[k][k]);
    for (int r = k + 1; r < 16; ++r) {
      float v = fabsf(Am[r][k]);
      if (v > best) { best = v; p = r; }
    }
    if (p != k) {
      for (int c = 0; c < 16; ++c) {
        float t = Am[k][c]; Am[k][c] = Am[p][c]; Am[p][c] = t;
      }
      float t = rhs[k]; rhs[k] = rhs[p]; rhs[p] = t;
    }
    float piv = Am[k][k];
    float inv = (fabsf(piv) > 1e-30f) ? 1.f / piv : 0.f;
    for (int r = k + 1; r < 16; ++r) {
      float f = Am[r][k] * inv;
      for (int c = k; c < 16; ++c) Am[r][c] -= f * Am[k][c];
      rhs[r] -= f * rhs[k];
    }
  }
  float x[16];
  for (int k = 15; k >= 0; --k) {
    float s = rhs[k];
    for (int c = k + 1; c < 16; ++c) s -= Am[k][c] * x[c];
    float d = Am[k][k];
    x[k] = (fabsf(d) > 1e-30f) ? s / d : 0.f;
  }
  for (int t = 0; t < 16; ++t) out[b * 16 + t] = x[t];
}

// ---------------------------------------------------------------------------
extern "C" void kernel_launch(void* const* d_in, const int* in_sizes, int n_in,
                              void* d_out, int out_size, void* d_ws,
                              size_t ws_size, hipStream_t stream) {
  (void)in_sizes; (void)n_in; (void)out_size;
  const float* q     = (const float*)d_in[0];
  const float* qdot  = (const float*)d_in[1];
  const float* W_in  = (const float*)d_in[2];
  const float* b_in  = (const float*)d_in[3];
  const float* W_h   = (const float*)d_in[4];
  const float* b_h   = (const float*)d_in[5];
  const float* W_out = (const float*)d_in[6];
  // b_out (d_in[7]) shifts L only; no effect on grad/Hessian.
  float* out = (float*)d_out;

  const float* W1 = W_h;
  const float* W2 = W_h + (size_t)H * H;
  const float* b1 = b_h;
  const float* b2 = b_h + H;

  float* p = (float*)d_ws;
  auto take = [&](size_t n) { float* r = p; p += n; return r; };
  const size_t HB = (size_t)H * BATCH;
  const size_t HN = (size_t)H * NCOLS;
  float* zT    = take(BATCH * ZD);
  float* x1T   = take(HB);
  float* x2T   = take(HB);
  float* sg1   = take(HB);
  float* sg2   = take(HB);
  float* sg3   = take(HB);
  float* u1T   = take(HB);
  float* u2T   = take(HB);
  float* s1T   = take(HB);
  float* s2T   = take(HB);
  float* s3T   = take(HB);
  float* gradT = take(BATCH * ZD);
  float* hvT   = take((size_t)BATCH * QD * ZD);
  float* slotA = take(HN);
  float* slotB = take(HN);
  float* slotC = take(HN);
  float* slotD = take(HN);
  if ((size_t)((char*)p - (char*)d_ws) > ws_size) return;  // ws too small

  const dim3 blk(128);
  const int ew = (int)(HN / 256);

  // ---- scalar forward/backward (N = 256) ----
  k_prep_z<<<dim3((BATCH * ZD) / 256), dim3(256), 0, stream>>>(q, qdot, zT);
  gemm_k<false, 1><<<dim3(BATCH / 64, H / 64), blk, 0, stream>>>(
      W_in, ZD, zT, x1T, sg1, b_in, nullptr, H, ZD, BATCH);
  gemm_k<false, 1><<<dim3(BATCH / 64, H / 64), blk, 0, stream>>>(
      W1, H, x1T, x2T, sg2, b1, nullptr, H, H, BATCH);
  gemm_k<false, 2><<<dim3(BATCH / 64, H / 64), blk, 0, stream>>>(
      W2, H, x2T, sg3, s3T, b2, W_out, H, H, BATCH);
  gemm_k<true, 3><<<dim3(BATCH / 64, H / 64), blk, 0, stream>>>(
      W2, H, s3T, u2T, s2T, nullptr, sg2, H, H, BATCH);
  gemm_k<true, 3><<<dim3(BATCH / 64, H / 64), blk, 0, stream>>>(
      W1, H, s2T, u1T, s1T, nullptr, sg1, H, H, BATCH);
  gemm_k<true, 0><<<dim3(BATCH / 64, 1), blk, 0, stream>>>(
      W_in, ZD, s1T, gradT, nullptr, nullptr, nullptr, ZD, H, BATCH);

  // ---- 16 HVPs per sample, chunked 64 samples (1024 columns) ----
  for (int ch = 0; ch < BATCH / CHUNK_B; ++ch) {
    int c0 = ch * CHUNK_B;
    k_dx1<<<ew, 256, 0, stream>>>(sg1, W_in, slotA, c0);                 // dx1
    gemm_k<false, 0><<<dim3(NCOLS / 64, H / 64), blk, 0, stream>>>(
        W1, H, slotA, slotB, nullptr, nullptr, nullptr, H, H, NCOLS);    // da2
    k_dx2<<<ew, 256, 0, stream>>>(sg2, slotB, slotC, c0);                // dx2
    gemm_k<false, 0><<<dim3(NCOLS / 64, H / 64), blk, 0, stream>>>(
        W2, H, slotC, slotD, nullptr, nullptr, nullptr, H, H, NCOLS);    // da3
    k_ds3<<<ew, 256, 0, stream>>>(sg3, W_out, slotD, c0);                // ds3
    gemm_k<true, 0><<<dim3(NCOLS / 64, H / 64), blk, 0, stream>>>(
        W2, H, slotD, slotA, nullptr, nullptr, nullptr, H, H, NCOLS);    // du2
    k_ds2<<<ew, 256, 0, stream>>>(sg2, u2T, slotA, slotB, c0);           // ds2
    gemm_k<true, 0><<<dim3(NCOLS / 64, H / 64), blk, 0, stream>>>(
        W1, H, slotB, slotC, nullptr, nullptr, nullptr, H, H, NCOLS);    // du1
    k_ds1<<<ew, 256, 0, stream>>>(sg1, u1T, W_in, slotC, c0);            // ds1
    gemm_k<true, 0><<<dim3(NCOLS / 64, 1), blk, 0, stream>>>(
        W_in, ZD, slotC, hvT + (size_t)c0 * QD * ZD, nullptr, nullptr,
        nullptr, ZD, H, NCOLS);                                          // Hv
  }

  k_solve<<<dim3(BATCH / 64), dim3(64), 0, stream>>>(hvT, gradT, qdot, out);
}